// GCE_74431783239746
// MI455X (gfx1250) — compile-verified
//
#include <hip/hip_runtime.h>
#include <hip/hip_bf16.h>

// exp(m) * (sparsemax(s) @ d),  N=500000, K=64, D=128, all fp32.
// Memory-bound (~386 MB @ 23.3 TB/s). Use full-precision f32 WMMA
// (V_WMMA_F32_16X16X4_F32) so the matmul rides the matrix pipe at no
// precision cost. Sparsemax via exact Michelot fixed-point per row.

typedef __attribute__((ext_vector_type(2))) float v2f;
typedef __attribute__((ext_vector_type(8))) float v8f;

#define ROWS_PER_BLOCK 128
#define TPB            128   // 4 waves (wave32)

__global__ __launch_bounds__(TPB)
void GCE_sparsemax_gemm_kernel(const float* __restrict__ m,
                               const float* __restrict__ s,
                               const float* __restrict__ dmat,
                               float* __restrict__ out,
                               int n)
{
    // p_lds: 128 rows x 64 cols f32, XOR-swizzled blocks of 4 floats:
    //   physical block = logical block ^ (row & 15)  -> conflict-free A loads.
    __shared__ float p_lds[ROWS_PER_BLOCK * 64];   // 32 KB
    __shared__ float d_lds[64 * 128];              // 32 KB, row-major (K x D)

    const int tid  = threadIdx.x;
    const int row0 = blockIdx.x * ROWS_PER_BLOCK;
    const int row  = row0 + tid;

    // ---- Stage d (K x D = 8192 floats) into LDS, fully coalesced ----
    {
        const float4* g4 = (const float4*)dmat;
        float4*       l4 = (float4*)d_lds;
        #pragma unroll
        for (int i = 0; i < 16; ++i)
            l4[tid + i * TPB] = g4[tid + i * TPB];
    }

    // ---- Per-thread sparsemax of one row (exact Michelot iteration) ----
    float z[64];
    if (row < n) {
        const float4* s4 = (const float4*)(s + (size_t)row * 64);
        #pragma unroll
        for (int i = 0; i < 16; ++i) {
            float4 v = s4[i];
            z[4*i+0] = v.x; z[4*i+1] = v.y; z[4*i+2] = v.z; z[4*i+3] = v.w;
        }
    } else {
        #pragma unroll
        for (int i = 0; i < 64; ++i) z[i] = 0.0f;
    }

    float sum = 0.0f;
    #pragma unroll
    for (int i = 0; i < 64; ++i) sum += z[i];
    int   cnt = 64;
    float tau = 0.0f;
    // tau = (sum_S - 1)/|S| over the shrinking support set; tau is
    // monotone increasing, so re-testing all 64 elements each pass is exact.
    for (int it = 0; it < 64; ++it) {
        tau = (sum - 1.0f) / (float)cnt;
        float ns = 0.0f; int nc = 0;
        #pragma unroll
        for (int i = 0; i < 64; ++i) {
            bool in = z[i] > tau;
            ns += in ? z[i] : 0.0f;
            nc += in ? 1 : 0;
        }
        if (nc == cnt) break;
        sum = ns; cnt = nc;
    }
    #pragma unroll
    for (int i = 0; i < 64; ++i) z[i] = fmaxf(z[i] - tau, 0.0f);

    // ---- Store p row into LDS with XOR block swizzle ----
    #pragma unroll
    for (int L = 0; L < 16; ++L) {
        int P = L ^ (tid & 15);
        *(float4*)&p_lds[tid * 64 + P * 4] =
            make_float4(z[4*L+0], z[4*L+1], z[4*L+2], z[4*L+3]);
    }
    __syncthreads();

    // ---- GEMM: wave w handles row tiles 2w and 2w+1 ----
    const int wave = tid >> 5;
    const int lane = tid & 31;
    const int hi   = lane >> 4;    // 0: lanes 0-15, 1: lanes 16-31
    const int l16  = lane & 15;

    // A fragments (16x4 f32 layout: VGPR0 = K0|K2, VGPR1 = K1|K3 per lane half)
    v2f   afrag[2][16];
    float em[2][8];
    #pragma unroll
    for (int t = 0; t < 2; ++t) {
        const int rbase = (wave * 2 + t) * 16;
        const int arow  = rbase + l16;           // arow & 15 == l16
        #pragma unroll
        for (int kk = 0; kk < 16; ++kk) {
            int P = kk ^ l16;                    // undo swizzle
            afrag[t][kk] = *(const v2f*)&p_lds[arow * 64 + P * 4 + hi * 2];
        }
        #pragma unroll
        for (int r = 0; r < 8; ++r) {
            int orow = row0 + rbase + r + hi * 8;
            float mv = (orow < n) ? m[orow] : 0.0f;
            em[t][r] = expf(mv);
        }
    }

    #pragma unroll
    for (int nt = 0; nt < 8; ++nt) {
        const int n0 = nt * 16;
        v8f c0 = {};
        v8f c1 = {};
        #pragma unroll
        for (int kk = 0; kk < 16; ++kk) {
            const int krow = kk * 4 + hi * 2;    // B 4x16: VGPR0 = K0|K2, VGPR1 = K1|K3
            v2f b;
            b[0] = d_lds[(krow    ) * 128 + n0 + l16];
            b[1] = d_lds[(krow + 1) * 128 + n0 + l16];
            c0 = __builtin_amdgcn_wmma_f32_16x16x4_f32(
                     false, afrag[0][kk], false, b, (short)0, c0, false, false);
            c1 = __builtin_amdgcn_wmma_f32_16x16x4_f32(
                     false, afrag[1][kk], false, b, (short)0, c1, false, false);
        }
        // C layout: VGPR r -> row (r | r+8), lane -> column. Scale by exp(m).
        #pragma unroll
        for (int r = 0; r < 8; ++r) {
            int orow0 = row0 + (wave * 2 + 0) * 16 + r + hi * 8;
            int orow1 = row0 + (wave * 2 + 1) * 16 + r + hi * 8;
            if (orow0 < n) out[(size_t)orow0 * 128 + n0 + l16] = em[0][r] * c0[r];
            if (orow1 < n) out[(size_t)orow1 * 128 + n0 + l16] = em[1][r] * c1[r];
        }
    }
}

extern "C" void kernel_launch(void* const* d_in, const int* in_sizes, int n_in,
                              void* d_out, int out_size, void* d_ws, size_t ws_size,
                              hipStream_t stream) {
    const float* m    = (const float*)d_in[0];   // [N,1]
    const float* s    = (const float*)d_in[1];   // [N,64]
    const float* dmat = (const float*)d_in[2];   // [64,128]
    float*       out  = (float*)d_out;           // [N,128]
    const int n = in_sizes[0];                   // N
    const int blocks = (n + ROWS_PER_BLOCK - 1) / ROWS_PER_BLOCK;
    GCE_sparsemax_gemm_kernel<<<dim3(blocks), dim3(TPB), 0, stream>>>(
        m, s, dmat, out, n);
}